// PG_FocalLossProb_77841987273461
// MI455X (gfx1250) — compile-verified
//
#include <hip/hip_runtime.h>
#include <hip/hip_bf16.h>
#include <math.h>

// Focal loss (prob-space) over [B=2048, V=32000] fp32 + sparse targets [B, T=20].
// DRAM-bound streaming reduction: 262 MB read -> ~11.2 us at 23.3 TB/s.
// Pass 1: one block per row, float4 (b128) coalesced streaming, negative-class
//         loss everywhere + sparse positive-class correction, LDS tree reduce.
// Pass 2: 2048 partials reduced by v_wmma_f32_16x16x4_f32 against an all-ones
//         B matrix (D[m][n] = sum_k A[m][k] + C[m][n]), one wave, deterministic.

#define FL_ALPHA 0.25f
#define FL_EPS   1e-8f

__device__ __forceinline__ float fl_neg_loss(float p) {
  // (1-ALPHA) * p^2 * (-ln(1-p)), with p clipped to [EPS, 1-EPS] in fp32.
  float pc = fminf(fmaxf(p, FL_EPS), 1.0f - FL_EPS);
  float l  = __logf(1.0f - pc);          // v_log_f32 path, natural log
  return -0.75f * pc * pc * l;
}

__device__ __forceinline__ float fl_pos_loss(float p) {
  // ALPHA * (1-p)^2 * (-ln(p))
  float pc = fminf(fmaxf(p, FL_EPS), 1.0f - FL_EPS);
  float q  = 1.0f - pc;
  float l  = __logf(pc);
  return -0.25f * q * q * l;
}

__global__ __launch_bounds__(256) void focal_row_kernel(
    const float* __restrict__ src, const int* __restrict__ tgt,
    float* __restrict__ partial, int V, int T) {
  const int b   = blockIdx.x;
  const int tid = threadIdx.x;
  const size_t rowoff = (size_t)b * (size_t)V;

  float acc = 0.0f;

  // Sparse positive-class corrections (threads 0..T-1), dedup duplicates
  // within the row (scatter .at[].set(1.0) is idempotent for repeats).
  if (tid < T) {
    const int* trow = tgt + (size_t)b * (size_t)T;
    int v = trow[tid];
    bool dup = false;
    for (int t = 0; t < tid; ++t) dup = dup || (trow[t] == v);
    if (!dup) {
      float p = src[rowoff + (size_t)v];
      acc += fl_pos_loss(p) - fl_neg_loss(p);
    }
  }

  // Streaming pass: V = 32000 -> 8000 float4 per row, fully coalesced b128.
  const float4* row4 = (const float4*)(src + rowoff);
  const int n4 = V >> 2;
  for (int i = tid; i < n4; i += 256) {
    float4 p = row4[i];
    acc += fl_neg_loss(p.x);
    acc += fl_neg_loss(p.y);
    acc += fl_neg_loss(p.z);
    acc += fl_neg_loss(p.w);
  }

  // Deterministic block reduction (fixed order, no float atomics).
  __shared__ float sdata[256];
  sdata[tid] = acc;
  __syncthreads();
  for (int s = 128; s > 0; s >>= 1) {
    if (tid < s) sdata[tid] += sdata[tid + s];
    __syncthreads();
  }
  if (tid == 0) partial[b] = sdata[0];
}

// ---- Final reduction via WMMA ------------------------------------------------
typedef __attribute__((ext_vector_type(2))) float v2f;
typedef __attribute__((ext_vector_type(8))) float v8f;

__global__ __launch_bounds__(32) void final_reduce_wmma(
    const float* __restrict__ partial, int n, float* __restrict__ out) {
  const int lane = threadIdx.x;  // one wave32, EXEC all ones for WMMA

  v8f c = {};                    // 16x16 f32 accumulator (8 VGPRs)
  v2f bones;                     // B = 4x16 all-ones (2 VGPRs)
  bones[0] = 1.0f;
  bones[1] = 1.0f;

  // Full 64-value chunks: lane L takes the contiguous pair {base+2L, base+2L+1}
  // as one unconditional b64 load. With B == 1: D[m][j] = sum_k A[m][k] + C[m][j];
  // any slot assignment of the 64 values into A is valid for a grand sum.
  const int nfull = n & ~63;
  for (int base = 0; base < nfull; base += 64) {
    v2f a = *(const v2f*)(partial + base + 2 * lane);
    c = __builtin_amdgcn_wmma_f32_16x16x4_f32(
        /*neg_a=*/false, a, /*neg_b=*/false, bones,
        /*c_mod=*/(short)0, c, /*reuse_a=*/false, /*reuse_b=*/false);
  }

  // Grand total = sum_m D[m][0]. Column 0 lives in lane 0 (M=0..7 across the
  // 8 accumulator VGPRs) and lane 16 (M=8..15).
  float sum8 = c[0] + c[1] + c[2] + c[3] + c[4] + c[5] + c[6] + c[7];
  float other = __shfl_xor(sum8, 16, 32);
  if (lane == 0) {
    float total = sum8 + other;
    for (int i = nfull; i < n; ++i) total += partial[i];  // scalar tail (empty for n=2048)
    out[0] = total;
  }
}

extern "C" void kernel_launch(void* const* d_in, const int* in_sizes, int n_in,
                              void* d_out, int out_size, void* d_ws, size_t ws_size,
                              hipStream_t stream) {
  const float* src = (const float*)d_in[0];   // [B, V] fp32 probabilities
  const int*   tgt = (const int*)d_in[1];     // [B, T] int32 token ids
  float* out     = (float*)d_out;             // scalar loss
  float* partial = (float*)d_ws;              // B per-row partials

  const int T  = 20;
  const int Bn = in_sizes[1] / T;             // 2048
  const int V  = in_sizes[0] / Bn;            // 32000

  focal_row_kernel<<<Bn, 256, 0, stream>>>(src, tgt, partial, V, T);
  final_reduce_wmma<<<1, 32, 0, stream>>>(partial, Bn, out);
}